// VanillaRNN_39797166965075
// MI455X (gfx1250) — compile-verified
//
#include <hip/hip_runtime.h>
#include <math.h>

typedef __attribute__((ext_vector_type(2))) float v2f;
typedef __attribute__((ext_vector_type(8))) float v8f;

#define RNN_HID 1024
#define RNN_BATCH 128
#define RNN_SEQ 512
#define NTILES 4   // 16x16 N-tiles per wave -> each wave computes a 16x64 strip

// Reorder row-major W(1024x1024) into K-pair interleaved layout:
//   WT[(k>>1)*2048 + n*2 + (k&1)] = W[k*1024 + n]
// so a lane's B fragment {W[k][n], W[k+1][n]} is one contiguous 8B load.
__global__ __launch_bounds__(256) void rnn_reorder_w_kernel(
    const float* __restrict__ W, float* __restrict__ WT)
{
    int idx = blockIdx.x * blockDim.x + threadIdx.x;   // 0 .. 1024*1024-1
    int k = idx >> 10;
    int n = idx & (RNN_HID - 1);
    WT[((size_t)(k >> 1) * RNN_HID + n) * 2 + (k & 1)] = W[idx];
}

// C(128x1024) = act( A0(128xK0) @ W0T [+ A1(128x1024) @ W1T] + bias )
// FUSE2: waves 0,1 split K of part0 (x @ W_xh), waves 2,3 split K of part1 (h @ W_hh)
// !FUSE2: all 4 waves split K=1024 of part0 (final h @ W_hy)
template <bool FUSE2, bool DO_TANH>
__global__ __launch_bounds__(128) void rnn_gemm_kernel(
    const float* __restrict__ A0, size_t a0_stride,
    const float* __restrict__ W0T,
    const float* __restrict__ A1,
    const float* __restrict__ W1T,
    const float* __restrict__ bias,
    float* __restrict__ out)
{
    const int lane = threadIdx.x & 31;
    const int wave = threadIdx.x >> 5;
    const int half = lane >> 4;   // K-pair select within the wave
    const int lr   = lane & 15;   // row (A) / col (B,C)

    const int n0 = blockIdx.x * (16 * NTILES);
    const int m0 = blockIdx.y * 16;

    const float* A;
    const float* WT;
    size_t astr;
    int kb, klen;
    if (FUSE2) {
        if (wave < 2) { A = A0; WT = W0T; astr = a0_stride; kb = wave * 512; }
        else          { A = A1; WT = W1T; astr = RNN_HID;   kb = (wave - 2) * 512; }
        klen = 512;
    } else {
        A = A0; WT = W0T; astr = a0_stride; kb = wave * 256; klen = 256;
    }

    v8f acc[NTILES] = {};

    // Pointer-increment form: constant strides per K-step so the backend can
    // use immediate offsets across the unroll and a single add per body.
    const float* ap = A + (size_t)(m0 + lr) * astr + (kb + half * 2);
    const float* wt = WT + ((size_t)((kb >> 1) + half) * RNN_HID + n0 + lr) * 2;

    for (int kk = 0; kk < klen; kk += 4) {
        v2f a = *(const v2f*)ap;           // A: K = kbase, kbase+1 (one b64)
#pragma unroll
        for (int i = 0; i < NTILES; i++) {
            v2f b = *(const v2f*)(wt + i * 32);   // B: one coalesced b64
            acc[i] = __builtin_amdgcn_wmma_f32_16x16x4_f32(
                false, a, false, b, (short)0, acc[i], false, false);
        }
        ap += 4;                           // +16 B
        wt += 2 * RNN_HID * 2;             // +16 KB (two K-pairs of WT)
    }

    // Reduce the 4 split-K partials across waves via LDS, then finalize.
    __shared__ float red[4][NTILES][8][32];
#pragma unroll
    for (int i = 0; i < NTILES; i++)
#pragma unroll
        for (int j = 0; j < 8; j++)
            red[wave][i][j][lane] = acc[i][j];
    __syncthreads();

    if (wave == 0) {
#pragma unroll
        for (int i = 0; i < NTILES; i++) {
            const int n = n0 + i * 16 + lr;
            const float bv = bias[n];
#pragma unroll
            for (int j = 0; j < 8; j++) {
                float v = red[0][i][j][lane] + red[1][i][j][lane] +
                          red[2][i][j][lane] + red[3][i][j][lane] + bv;
                if (DO_TANH) v = tanhf(v);
                // C/D layout: VGPR j, lanes 0-15 -> M=j ; lanes 16-31 -> M=j+8
                out[(size_t)(m0 + j + half * 8) * RNN_HID + n] = v;
            }
        }
    }
}

__global__ void rnn_zero_kernel(float* __restrict__ p, int n)
{
    int i = blockIdx.x * blockDim.x + threadIdx.x;
    if (i < n) p[i] = 0.0f;
}

extern "C" void kernel_launch(void* const* d_in, const int* in_sizes, int n_in,
                              void* d_out, int out_size, void* d_ws, size_t ws_size,
                              hipStream_t stream)
{
    (void)in_sizes; (void)n_in; (void)out_size; (void)ws_size;
    const float* x    = (const float*)d_in[0];  // (128, 512, 1024)
    const float* W_xh = (const float*)d_in[1];  // (1024, 1024)
    const float* W_hh = (const float*)d_in[2];  // (1024, 1024)
    const float* b_h  = (const float*)d_in[3];  // (1024,)
    const float* W_hy = (const float*)d_in[4];  // (1024, 1024)
    const float* b_y  = (const float*)d_in[5];  // (1024,)
    float* out = (float*)d_out;                 // (128, 1024)

    const int hElems = RNN_BATCH * RNN_HID;     // 128*1024
    const int wElems = RNN_HID * RNN_HID;       // 1024*1024

    // Workspace layout: h ping/pong (1 MB) + 3 reordered weight matrices (12 MB)
    float* hA    = (float*)d_ws;
    float* hB    = hA + hElems;
    float* WTxh  = hB + hElems;
    float* WThh  = WTxh + wElems;
    float* WThy  = WThh + wElems;

    // h0 = zeros (ws is poisoned by the harness; re-init every call)
    rnn_zero_kernel<<<hElems / 256, 256, 0, stream>>>(hA, hElems);

    // One-time (per call) weight reorder; amortized over 512 step GEMMs,
    // and keeps all weights L2-resident (12 MB << 192 MB).
    rnn_reorder_w_kernel<<<wElems / 256, 256, 0, stream>>>(W_xh, WTxh);
    rnn_reorder_w_kernel<<<wElems / 256, 256, 0, stream>>>(W_hh, WThh);
    rnn_reorder_w_kernel<<<wElems / 256, 256, 0, stream>>>(W_hy, WThy);

    dim3 grid(RNN_HID / (16 * NTILES), RNN_BATCH / 16);  // (16, 8)
    dim3 block(128);                                     // 4 waves (wave32)

    float* hin = hA;
    float* hout = hB;
    for (int t = 0; t < RNN_SEQ; t++) {
        // x_t rows: x[b][t][:] -> base x + t*1024, row stride 512*1024
        rnn_gemm_kernel<true, true><<<grid, block, 0, stream>>>(
            x + (size_t)t * RNN_HID, (size_t)RNN_SEQ * RNN_HID, WTxh,
            hin, WThh, b_h, hout);
        float* tmp = hin; hin = hout; hout = tmp;
    }

    // y = h_final @ W_hy + b_y (no tanh), straight to d_out
    rnn_gemm_kernel<false, false><<<grid, block, 0, stream>>>(
        hin, (size_t)RNN_HID, WThy, nullptr, nullptr, b_y, out);
}